// PrototypicalTextClassifier_4861902979756
// MI455X (gfx1250) — compile-verified
//
#include <hip/hip_runtime.h>
#include <hip/hip_bf16.h>

typedef __attribute__((ext_vector_type(2))) float v2f;
typedef __attribute__((ext_vector_type(8))) float v8f;

#define DMODEL 512
#define NTOK   64
#define NCLS   64

// ---------------------------------------------------------------------------
// Zero-init a float buffer (proto sums + counts).
// ---------------------------------------------------------------------------
__global__ void zero_kernel(float* __restrict__ p, int n) {
    int i = blockIdx.x * blockDim.x + threadIdx.x;
    if (i < n) p[i] = 0.0f;
}

// ---------------------------------------------------------------------------
// Embedding gather + mean pool for query rows; also emits ||q||^2 per row.
// One block (256 thr) per row; thread t owns dims t and t+256 (coalesced).
// Table (64MB) fits in the 192MB L2, so these gathers mostly hit L2.
// ---------------------------------------------------------------------------
__global__ void pool_query_kernel(const int* __restrict__ tokens,   // [Q, 64]
                                  const float* __restrict__ table,  // [V, 512]
                                  float* __restrict__ out,          // [Q, 512]
                                  float* __restrict__ out_sq)       // [Q]
{
    const int row = blockIdx.x;
    const int t   = threadIdx.x;                 // 0..255
    const int* tok = tokens + row * NTOK;
    float a0 = 0.0f, a1 = 0.0f;
    #pragma unroll 8
    for (int i = 0; i < NTOK; ++i) {
        const float* e = table + (size_t)tok[i] * DMODEL;
        a0 += e[t];
        a1 += e[t + 256];
    }
    a0 *= (1.0f / NTOK);
    a1 *= (1.0f / NTOK);
    out[(size_t)row * DMODEL + t]       = a0;
    out[(size_t)row * DMODEL + t + 256] = a1;

    __shared__ float red[256];
    red[t] = a0 * a0 + a1 * a1;
    __syncthreads();
    for (int s = 128; s > 0; s >>= 1) {
        if (t < s) red[t] += red[t + s];
        __syncthreads();
    }
    if (t == 0) out_sq[row] = red[0];
}

// ---------------------------------------------------------------------------
// Support rows: gather + mean pool, then atomic segment-sum into per-class
// prototype sums (float atomics resolved in L2) + class counts.
// ---------------------------------------------------------------------------
__global__ void support_proto_kernel(const int* __restrict__ tokens,   // [S, 64]
                                     const int* __restrict__ labels,   // [S]
                                     const float* __restrict__ table,  // [V, 512]
                                     float* __restrict__ protoSum,     // [C, 512]
                                     float* __restrict__ counts)       // [C]
{
    const int row = blockIdx.x;
    const int t   = threadIdx.x;
    const int* tok = tokens + row * NTOK;
    float a0 = 0.0f, a1 = 0.0f;
    #pragma unroll 8
    for (int i = 0; i < NTOK; ++i) {
        const float* e = table + (size_t)tok[i] * DMODEL;
        a0 += e[t];
        a1 += e[t + 256];
    }
    a0 *= (1.0f / NTOK);
    a1 *= (1.0f / NTOK);
    const int lab = labels[row];
    atomicAdd(&protoSum[(size_t)lab * DMODEL + t],       a0);
    atomicAdd(&protoSum[(size_t)lab * DMODEL + t + 256], a1);
    if (t == 0) atomicAdd(&counts[lab], 1.0f);
}

// ---------------------------------------------------------------------------
// protos = sums / max(count,1); p2[c] = ||proto_c||^2. One block per class.
// ---------------------------------------------------------------------------
__global__ void finalize_proto_kernel(const float* __restrict__ protoSum,
                                      const float* __restrict__ counts,
                                      float* __restrict__ protos,  // [C, 512]
                                      float* __restrict__ p2)      // [C]
{
    const int c = blockIdx.x;
    const int t = threadIdx.x;
    const float inv = 1.0f / fmaxf(counts[c], 1.0f);
    float v0 = protoSum[(size_t)c * DMODEL + t]       * inv;
    float v1 = protoSum[(size_t)c * DMODEL + t + 256] * inv;
    protos[(size_t)c * DMODEL + t]       = v0;
    protos[(size_t)c * DMODEL + t + 256] = v1;

    __shared__ float red[256];
    red[t] = v0 * v0 + v1 * v1;
    __syncthreads();
    for (int s = 128; s > 0; s >>= 1) {
        if (t < s) red[t] += red[t + s];
        __syncthreads();
    }
    if (t == 0) p2[c] = red[0];
}

// ---------------------------------------------------------------------------
// Distance: out[m][n] = 2*(q.p) - ||q||^2 - ||p||^2 via V_WMMA_F32_16X16X4_F32.
// One wave32 per 16x16 output tile; K swept 512/4 = 128 WMMA ops.
// A layout: lane=(half<<4)|m, vgpr j -> K = 2*half + j.
// B layout: lane=(half<<4)|n, vgpr j -> K = 2*half + j (B[k][n] = P[n][k]).
// C/D layout: vgpr i -> M = i + 8*half, lane&15 -> N.
// ---------------------------------------------------------------------------
__global__ void dist_wmma_kernel(const float* __restrict__ q,   // [Q, 512]
                                 const float* __restrict__ q2,  // [Q]
                                 const float* __restrict__ p,   // [C, 512]
                                 const float* __restrict__ p2,  // [C]
                                 float* __restrict__ out)       // [Q, C]
{
    const int lane = threadIdx.x;        // 0..31, full wave, no divergence
    const int half = lane >> 4;
    const int lidx = lane & 15;
    const int m0 = blockIdx.y * 16;      // query tile
    const int n0 = blockIdx.x * 16;      // class tile

    const float* qrow = q + (size_t)(m0 + lidx) * DMODEL + 2 * half;
    const float* prow = p + (size_t)(n0 + lidx) * DMODEL + 2 * half;

    v8f acc = {};
    #pragma unroll 8
    for (int k = 0; k < DMODEL; k += 4) {
        v2f a = *(const v2f*)(qrow + k);
        v2f b = *(const v2f*)(prow + k);
        acc = __builtin_amdgcn_wmma_f32_16x16x4_f32(
            /*neg_a=*/false, a, /*neg_b=*/false, b,
            /*c_mod=*/(short)0, acc, /*reuse_a=*/false, /*reuse_b=*/false);
    }

    const float p2v = p2[n0 + lidx];
    #pragma unroll
    for (int i = 0; i < 8; ++i) {
        const int m = m0 + i + 8 * half;
        out[(size_t)m * NCLS + n0 + lidx] = 2.0f * acc[i] - q2[m] - p2v;
    }
}

// ---------------------------------------------------------------------------
extern "C" void kernel_launch(void* const* d_in, const int* in_sizes, int n_in,
                              void* d_out, int out_size, void* d_ws, size_t ws_size,
                              hipStream_t stream) {
    const int*   query   = (const int*)d_in[0];     // [Q, 64]
    const int*   support = (const int*)d_in[1];     // [S, 64]
    const int*   labels  = (const int*)d_in[2];     // [S]
    const float* table   = (const float*)d_in[3];   // [V, 512]
    float*       out     = (float*)d_out;           // [Q, 64]

    const int Q = in_sizes[0] / NTOK;               // 2048
    const int S = in_sizes[2];                      // 4096

    // Workspace layout (floats)
    float* ws       = (float*)d_ws;
    float* qEmb     = ws;                                   // Q*512
    float* qSq      = qEmb     + (size_t)Q * DMODEL;        // Q
    float* protoSum = qSq      + Q;                         // 64*512
    float* counts   = protoSum + NCLS * DMODEL;             // 64
    float* protos   = counts   + NCLS;                      // 64*512
    float* pSq      = protos   + NCLS * DMODEL;             // 64

    // 1) zero proto sums + counts
    const int nz = NCLS * DMODEL + NCLS;
    zero_kernel<<<(nz + 255) / 256, 256, 0, stream>>>(protoSum, nz);

    // 2) query embeddings (+ squared norms)
    pool_query_kernel<<<Q, 256, 0, stream>>>(query, table, qEmb, qSq);

    // 3) support embeddings -> atomic segment sums
    support_proto_kernel<<<S, 256, 0, stream>>>(support, labels, table,
                                                protoSum, counts);

    // 4) prototypes + squared norms
    finalize_proto_kernel<<<NCLS, 256, 0, stream>>>(protoSum, counts, protos, pSq);

    // 5) fp32 WMMA distance: grid (C/16, Q/16), one wave per tile
    dim3 grid(NCLS / 16, Q / 16);
    dist_wmma_kernel<<<grid, 32, 0, stream>>>(qEmb, qSq, protos, pSq, out);
}